// ColorHistogramLoss_52553219834464
// MI455X (gfx1250) — compile-verified
//
#include <hip/hip_runtime.h>
#include <stdint.h>

#define BINS 64
#define SLICE_ELEMS (512 * 512)            // floats per (b,c) slice
#define SLICES_PER_TENSOR 96               // 32*3
#define WGS_PER_SLICE 16
#define THREADS 256
#define WAVES 8                            // 256 / wave32
#define V4_PER_SLICE (SLICE_ELEMS / 4)     // 65536 float4
#define V4_PER_WG (V4_PER_SLICE / WGS_PER_SLICE)   // 4096
#define V4_PER_WAVE (V4_PER_WG / WAVES)            // 512
#define STAGES (V4_PER_WAVE / 32)                  // 16 pipeline stages per wave
#define TOTAL_HIST (2 * SLICES_PER_TENSOR * BINS)  // 12288 u32 counters in d_ws

// ---- CDNA5 async global->LDS copy (ASYNCcnt path), inline asm ----
__device__ __forceinline__ void async_load_b128(uint32_t lds_byte_off, const void* gaddr) {
    // VDST = per-lane LDS byte offset, VADDR = per-lane 64-bit global address (GV mode)
    asm volatile("global_load_async_to_lds_b128 %0, %1, off"
                 :
                 : "v"(lds_byte_off), "v"(gaddr)
                 : "memory");
}
#define WAIT_ASYNC_1() asm volatile("s_wait_asynccnt 1" ::: "memory")
#define WAIT_ASYNC_0() asm volatile("s_wait_asynccnt 0" ::: "memory")

__global__ void __launch_bounds__(THREADS) zero_hist(uint32_t* __restrict__ h, int n) {
    int i = blockIdx.x * blockDim.x + threadIdx.x;
    if (i < n) h[i] = 0u;
}

__device__ __forceinline__ void bin_one(float v, uint32_t* __restrict__ h) {
    // edges = linspace(-1,1,64); idx = searchsorted(right)-1 == floor((v+1)*31.5)
    // idx < 0  -> invalid (v < -1), no count;  idx > 63 -> clamp (open last bin)
    int idx = (int)floorf((v + 1.0f) * 31.5f);
    if (idx >= 0) {
        idx = idx > (BINS - 1) ? (BINS - 1) : idx;
        atomicAdd(&h[idx], 1u);   // ds_add_u32, wave-private histogram
    }
}

__global__ void __launch_bounds__(THREADS) hist_kernel(const float* __restrict__ in,
                                                       const float* __restrict__ tgt,
                                                       uint32_t* __restrict__ ghist) {
    __shared__ float4   stage[WAVES][2][32];   // double-buffered 512B per wave
    __shared__ uint32_t hist[WAVES][BINS];     // wave-privatized histograms

    const int tid  = threadIdx.x;
    const int lane = tid & 31;
    const int wave = tid >> 5;

    // zero own wave's histogram (wave-coherent: DS ops in-order within a wave)
    hist[wave][lane]      = 0u;
    hist[wave][lane + 32] = 0u;

    const int g           = blockIdx.x;            // 0 .. 192*16-1
    const int slice       = g / WGS_PER_SLICE;     // 0..191 (<96: input, >=96: target)
    const int wg_in_slice = g - slice * WGS_PER_SLICE;
    const bool is_in      = slice < SLICES_PER_TENSOR;
    const int  s          = is_in ? slice : (slice - SLICES_PER_TENSOR);

    const float4* base = (const float4*)(is_in ? in : tgt)
                         + (size_t)s * V4_PER_SLICE
                         + (size_t)wg_in_slice * V4_PER_WG
                         + (size_t)wave * V4_PER_WAVE;

    // Per-lane LDS byte offsets for the two staging slots (low 32 bits of the
    // generic pointer to a __shared__ object == LDS byte offset, ISA 10.2).
    const uint32_t lds0 = (uint32_t)(uintptr_t)&stage[wave][0][lane];
    const uint32_t lds1 = (uint32_t)(uintptr_t)&stage[wave][1][lane];

    // prime the pipeline: stage 0
    async_load_b128(lds0, (const void*)(base + lane));

    for (int i = 0; i < STAGES; ++i) {
        if (i + 1 < STAGES) {
            const uint32_t nxt_off = ((i + 1) & 1) ? lds1 : lds0;
            async_load_b128(nxt_off, (const void*)(base + (i + 1) * 32 + lane));
            WAIT_ASYNC_1();   // async loads complete in order -> stage i is ready
        } else {
            WAIT_ASYNC_0();
        }
        const float4 v = stage[wave][i & 1][lane];
        bin_one(v.x, hist[wave]);
        bin_one(v.y, hist[wave]);
        bin_one(v.z, hist[wave]);
        bin_one(v.w, hist[wave]);
    }

    __syncthreads();
    if (tid < BINS) {
        uint32_t sum = 0;
        #pragma unroll
        for (int w = 0; w < WAVES; ++w) sum += hist[w][tid];
        atomicAdd(&ghist[(size_t)slice * BINS + tid], sum);  // global_atomic_add_u32
    }
}

__global__ void __launch_bounds__(THREADS) loss_kernel(const uint32_t* __restrict__ ghist,
                                                       float* __restrict__ out) {
    __shared__ uint32_t psum[THREADS];
    uint32_t acc = 0;
    for (int i = threadIdx.x; i < SLICES_PER_TENSOR * BINS; i += THREADS) {
        int a = (int)ghist[i];
        int b = (int)ghist[SLICES_PER_TENSOR * BINS + i];
        int d = a - b;
        acc += (uint32_t)(d < 0 ? -d : d);
    }
    psum[threadIdx.x] = acc;
    __syncthreads();
    for (int s = THREADS / 2; s > 0; s >>= 1) {
        if (threadIdx.x < (unsigned)s) psum[threadIdx.x] += psum[threadIdx.x + s];
        __syncthreads();
    }
    if (threadIdx.x == 0) {
        const double N = (double)SLICE_ELEMS;
        const double M = (double)(SLICES_PER_TENSOR * BINS);
        out[0] = (float)((double)psum[0] / (N * M));
    }
}

extern "C" void kernel_launch(void* const* d_in, const int* in_sizes, int n_in,
                              void* d_out, int out_size, void* d_ws, size_t ws_size,
                              hipStream_t stream) {
    const float* in  = (const float*)d_in[0];
    const float* tgt = (const float*)d_in[1];
    uint32_t* ghist  = (uint32_t*)d_ws;     // 192 * 64 u32 = 48 KB
    float* out       = (float*)d_out;

    zero_hist<<<(TOTAL_HIST + THREADS - 1) / THREADS, THREADS, 0, stream>>>(ghist, TOTAL_HIST);
    hist_kernel<<<2 * SLICES_PER_TENSOR * WGS_PER_SLICE, THREADS, 0, stream>>>(in, tgt, ghist);
    loss_kernel<<<1, THREADS, 0, stream>>>(ghist, out);
}